// M7_66013647339907
// MI455X (gfx1250) — compile-verified
//
#include <hip/hip_runtime.h>

typedef __attribute__((ext_vector_type(2))) float v2f;
typedef __attribute__((ext_vector_type(8))) float v8f;
typedef __attribute__((ext_vector_type(4))) int v4i;

#define N_NODES 50000
#define N_EDGES 800000
#define N_GRAPHS 2000
#define DN 64
#define DM 128

__device__ __forceinline__ v8f wmma4(v2f a, v2f b, v8f c) {
  // V_WMMA_F32_16X16X4_F32 : D = A(16x4) * B(4x16) + C(16x16), fp32
  return __builtin_amdgcn_wmma_f32_16x16x4_f32(false, a, false, b, (short)0, c,
                                               false, false);
}

__device__ __forceinline__ float lrelu(float v) {
  return v >= 0.0f ? v : 0.01f * v;
}

// Stage one 16-byte segment from gsrc into LDS via the CDNA5 async
// global->LDS path (ASYNCcnt) when available; plain store fallback otherwise.
__device__ __forceinline__ void stage_to_lds(float* lds, const float* gsrc,
                                             int seg_idx, bool active) {
#if __has_builtin(__builtin_amdgcn_global_load_async_to_lds_b128)
  if (active) {
    __builtin_amdgcn_global_load_async_to_lds_b128(
        (__attribute__((address_space(1))) v4i*)gsrc,
        (__attribute__((address_space(3))) v4i*)(lds + seg_idx * 4), 0, 0);
  }
#if __has_builtin(__builtin_amdgcn_s_wait_asynccnt)
  __builtin_amdgcn_s_wait_asynccnt(0);
#else
  asm volatile("s_wait_asynccnt 0" ::: "memory");
#endif
#else
  if (active) {
    lds[seg_idx * 4 + 0] = gsrc[0];
    lds[seg_idx * 4 + 1] = gsrc[1];
    lds[seg_idx * 4 + 2] = gsrc[2];
    lds[seg_idx * 4 + 3] = gsrc[3];
  }
#endif
}

// ---------------------------------------------------------------------------
// Edge kernel: one wave handles 16 edges.
//   msg = relu(edge_attr[16x16] @ We[16x64] + be + x[src])
//   aggr[dst] += msg  (global fp32 atomics, L2-resident)
// We (16x64 = 4KB) is staged in LDS once per block (async global->LDS).
// ---------------------------------------------------------------------------
__global__ void gin_edge_kernel(const float* __restrict__ x,
                                const float* __restrict__ ea,
                                const float* __restrict__ We,
                                const float* __restrict__ be,
                                const int* __restrict__ src,
                                const int* __restrict__ dst,
                                float* __restrict__ aggr, int nedges) {
  __shared__ float lds_we[16 * DN];  // 4 KB
  const int tid = threadIdx.x;
  // 16 rows x 16 segs(4 floats) = 256 segs, one per thread
  stage_to_lds(lds_we, We + tid * 4, tid, true);
  __syncthreads();

  const int wave = (int)((blockIdx.x * (long)blockDim.x + tid) >> 5);
  const int e0 = wave * 16;
  if (e0 >= nedges) return;
  const int lane = tid & 31;
  const int mloc = lane & 15;
  const int hi = lane >> 4;
  const int erow = e0 + mloc;

  v8f acc[4];
#pragma unroll
  for (int ct = 0; ct < 4; ++ct) acc[ct] = {};

#pragma unroll
  for (int k = 0; k < 16; k += 4) {
    const int kk = k + hi * 2;
    v2f a;
    a.x = ea[erow * 16 + kk];
    a.y = ea[erow * 16 + kk + 1];
#pragma unroll
    for (int ct = 0; ct < 4; ++ct) {
      const int col = ct * 16 + mloc;
      v2f b;
      b.x = lds_we[kk * DN + col];
      b.y = lds_we[(kk + 1) * DN + col];
      acc[ct] = wmma4(a, b, acc[ct]);
    }
  }

  float bvv[4];
#pragma unroll
  for (int ct = 0; ct < 4; ++ct) bvv[ct] = be[ct * 16 + mloc];

  // Per-lane output rows: edge m = e0 + hi*8 + r, r = 0..7
#pragma unroll
  for (int r = 0; r < 8; ++r) {
    const int m = e0 + hi * 8 + r;
    const float* xp = x + (size_t)src[m] * DN + mloc;
    float* ap = aggr + (size_t)dst[m] * DN + mloc;
#pragma unroll
    for (int ct = 0; ct < 4; ++ct) {
      float v = acc[ct][r] + bvv[ct] + xp[ct * 16];
      v = v > 0.0f ? v : 0.0f;  // relu
      atomicAdd(&ap[ct * 16], v);
    }
  }
}

// ---------------------------------------------------------------------------
// Fused GEMM: out[nrows x C] = f(A) @ W[K x C] + bias.
// One wave per 16x16 output tile; all 8 waves of a block share one column
// tile, whose W[K x 16] slab is staged in LDS once (async global->LDS).
// MODEA 0: a = A
// MODEA 1: a = (1+*eps_p)*A + A2                      (GIN combine)
// MODEA 2: a = lrelu(A*pre[k] + pre[K+k])             (BN-affine + leakyReLU)
// STATS : atomically accumulate column sum / sumsq of the (pre-act) out.
// POSTL : apply leaky-ReLU before storing (only used when !STATS).
// ---------------------------------------------------------------------------
template <int MODEA, bool POSTL, bool STATS>
__global__ void gemm_fused(const float* __restrict__ A,
                           const float* __restrict__ A2,
                           const float* __restrict__ eps_p,
                           const float* __restrict__ pre,
                           const float* __restrict__ W,
                           const float* __restrict__ bias,
                           float* __restrict__ out, float* __restrict__ stats,
                           int nrows, int K, int C) {
  __shared__ float lds_w[DM * 16];  // up to 8 KB (K<=128)
  const int tid = threadIdx.x;
  const int ctiles = C >> 4;
  const int ct = blockIdx.x % ctiles;
  const int rtg = blockIdx.x / ctiles;

  // Stage W[:, ct*16 : ct*16+16] -> lds_w[k*16 + n]; K*4 b128 segments.
  for (int i = tid; i < K * 4; i += 256) {
    const int k = i >> 2;
    const int seg = i & 3;
    stage_to_lds(lds_w, W + (size_t)k * C + ct * 16 + seg * 4, k * 4 + seg,
                 true);
  }
  __syncthreads();

  const int rt = rtg * 8 + (tid >> 5);
  if (rt * 16 >= nrows) return;
  const int lane = tid & 31;
  const int mloc = lane & 15;
  const int hi = lane >> 4;
  const int arow = rt * 16 + mloc;
  const int col = ct * 16 + mloc;

  float epsf = 1.0f;
  if (MODEA == 1) epsf = 1.0f + eps_p[0];

  v8f acc = {};
  for (int k = 0; k < K; k += 4) {
    const int kk = k + hi * 2;
    v2f a, b;
#pragma unroll
    for (int r = 0; r < 2; ++r) {
      float av = A[(long)arow * K + kk + r];
      if (MODEA == 1) av = epsf * av + A2[(long)arow * K + kk + r];
      if (MODEA == 2) av = lrelu(av * pre[kk + r] + pre[K + kk + r]);
      a[r] = av;
      b[r] = lds_w[(kk + r) * 16 + mloc];
    }
    acc = wmma4(a, b, acc);
  }

  const float bv = bias ? bias[col] : 0.0f;
  float s = 0.0f, s2 = 0.0f;
  float* op = out + (long)(rt * 16 + hi * 8) * C + col;
#pragma unroll
  for (int r = 0; r < 8; ++r) {
    float v = acc[r] + bv;
    if (POSTL) v = lrelu(v);
    op[r * C] = v;
    if (STATS) {
      s += v;
      s2 += v * v;
    }
  }
  if (STATS) {
    atomicAdd(&stats[col], s);
    atomicAdd(&stats[C + col], s2);
  }
}

// stats(sum,sumsq) + gamma/beta -> folded scale/bias:  y = h*sAB[c] + sAB[C+c]
__global__ void bn_finalize(const float* __restrict__ stats,
                            const float* __restrict__ g,
                            const float* __restrict__ b, float invn,
                            float* __restrict__ sAB, int C) {
  const int c = threadIdx.x;
  if (c >= C) return;
  const float m = stats[c] * invn;
  const float var = stats[C + c] * invn - m * m;
  const float inv = rsqrtf(var + 1e-5f);
  const float sa = g[c] * inv;
  sAB[c] = sa;
  sAB[C + c] = b[c] - m * sa;
}

__global__ void bn_act_kernel(const float* __restrict__ h,
                              const float* __restrict__ sAB,
                              float* __restrict__ out, int total, int C) {
  const int i = blockIdx.x * blockDim.x + threadIdx.x;
  if (i >= total) return;
  const int c = i & (C - 1);
  out[i] = lrelu(h[i] * sAB[c] + sAB[C + c]);
}

__global__ void pool_kernel(const float* __restrict__ x,
                            const int* __restrict__ batch,
                            float* __restrict__ pool, int total) {
  const int i = blockIdx.x * blockDim.x + threadIdx.x;
  if (i >= total) return;
  const int row = i >> 6;
  const int c = i & 63;
  atomicAdd(&pool[(long)batch[row] * DN + c], x[i]);
}

__global__ void hcat_kernel(const float* __restrict__ x,
                            const float* __restrict__ pool,
                            const int* __restrict__ batch,
                            float* __restrict__ hcat, int total) {
  const int i = blockIdx.x * blockDim.x + threadIdx.x;
  if (i >= total) return;
  const int row = i >> 7;
  const int c = i & 127;
  hcat[i] =
      (c < DN) ? x[(long)row * DN + c] : pool[(long)batch[row] * DN + (c - DN)];
}

// out[row] = sum_c lrelu(m2[row][c]*sA[c]+sB[c]) * cW3[c] + cb3  (wave per row)
__global__ void final_dot(const float* __restrict__ m2,
                          const float* __restrict__ sAB,
                          const float* __restrict__ cW3,
                          const float* __restrict__ cb3,
                          float* __restrict__ out, int nrows) {
  const int wave = (int)((blockIdx.x * (long)blockDim.x + threadIdx.x) >> 5);
  const int lane = threadIdx.x & 31;
  if (wave >= nrows) return;
  float s = 0.0f;
#pragma unroll
  for (int c = lane; c < DM; c += 32) {
    const float v = lrelu(m2[(long)wave * DM + c] * sAB[c] + sAB[DM + c]);
    s += v * cW3[c];
  }
#pragma unroll
  for (int off = 16; off > 0; off >>= 1) s += __shfl_xor(s, off, 32);
  if (lane == 0) out[wave] = s + cb3[0];
}

// ---------------------------------------------------------------------------
extern "C" void kernel_launch(void* const* d_in, const int* in_sizes, int n_in,
                              void* d_out, int out_size, void* d_ws,
                              size_t ws_size, hipStream_t stream) {
  const float* x_in = (const float*)d_in[0];
  const float* edge_attr = (const float*)d_in[1];
  const float* leW = (const float*)d_in[2];
  const float* leB = (const float*)d_in[3];
  const float* W1 = (const float*)d_in[4];
  const float* b1 = (const float*)d_in[5];
  const float* bn1g = (const float*)d_in[6];
  const float* bn1b = (const float*)d_in[7];
  const float* W2 = (const float*)d_in[8];
  const float* b2 = (const float*)d_in[9];
  const float* eps = (const float*)d_in[10];
  const float* obng = (const float*)d_in[11];
  const float* obnb = (const float*)d_in[12];
  const float* cW1 = (const float*)d_in[13];
  const float* cb1 = (const float*)d_in[14];
  const float* cbn1g = (const float*)d_in[15];
  const float* cbn1b = (const float*)d_in[16];
  const float* cW2 = (const float*)d_in[17];
  const float* cb2 = (const float*)d_in[18];
  const float* cbn2g = (const float*)d_in[19];
  const float* cbn2b = (const float*)d_in[20];
  const float* cW3 = (const float*)d_in[21];
  const float* cb3 = (const float*)d_in[22];
  const int* ei = (const int*)d_in[23];
  const int* batch = (const int*)d_in[24];
  const int* src = ei;
  const int* dst = ei + N_EDGES;

  float* ws = (float*)d_ws;
  float* x_cur = ws;                           // N*DN
  float* bufA = x_cur + (size_t)N_NODES * DN;  // N*DM : aggr/h2, then hcat/m2
  float* bufB = bufA + (size_t)N_NODES * DM;   // N*DM : t, then m1
  float* pool = bufB + (size_t)N_NODES * DM;   // G*DN
  float* stats = pool + (size_t)N_GRAPHS * DN; // 256
  float* sAB = stats + 256;                    // 256
  float* sAB2 = sAB + 256;                     // 256

  const float invn = 1.0f / (float)N_NODES;
  const dim3 blk(256);
  const int rtiles = N_NODES / 16;             // 3125
  const int rgroups = (rtiles + 7) / 8;        // 391

  const dim3 g_edge((unsigned)((N_EDGES / 16 * 32 + 255) / 256));  // 6250
  const dim3 g_gemm64((unsigned)(rgroups * 4));                    // 1564
  const dim3 g_gemm128((unsigned)(rgroups * 8));                   // 3128
  const dim3 g_elem64((N_NODES * DN + 255) / 256);
  const dim3 g_elem128((N_NODES * DM + 255) / 256);
  const dim3 g_dot((unsigned)((N_NODES * 32 + 255) / 256));

  for (int l = 0; l < 3; ++l) {
    const float* xl = (l == 0) ? x_in : x_cur;
    // aggr = segment_sum(relu(x[src] + edge_attr@We + be), dst)
    (void)hipMemsetAsync(bufA, 0, (size_t)N_NODES * DN * sizeof(float), stream);
    gin_edge_kernel<<<g_edge, blk, 0, stream>>>(
        xl, edge_attr, leW + (size_t)l * 16 * DN, leB + (size_t)l * DN, src,
        dst, bufA, N_EDGES);
    // t = ((1+eps)*x + aggr) @ W1 + b1 ; accumulate BN1 stats
    (void)hipMemsetAsync(stats, 0, 2 * DN * sizeof(float), stream);
    gemm_fused<1, false, true><<<g_gemm64, blk, 0, stream>>>(
        xl, bufA, eps + l, nullptr, W1 + (size_t)l * DN * DN,
        b1 + (size_t)l * DN, bufB, stats, N_NODES, DN, DN);
    bn_finalize<<<1, DN, 0, stream>>>(stats, bn1g + (size_t)l * DN,
                                      bn1b + (size_t)l * DN, invn, sAB, DN);
    if (l < 2) {
      // h2 = lrelu(bn1(t)) @ W2 + b2 ; accumulate output-BN stats
      (void)hipMemsetAsync(stats, 0, 2 * DN * sizeof(float), stream);
      gemm_fused<2, false, true><<<g_gemm64, blk, 0, stream>>>(
          bufB, nullptr, nullptr, sAB, W2 + (size_t)l * DN * DN,
          b2 + (size_t)l * DN, bufA, stats, N_NODES, DN, DN);
      bn_finalize<<<1, DN, 0, stream>>>(stats, obng + (size_t)l * DN,
                                        obnb + (size_t)l * DN, invn, sAB2, DN);
      bn_act_kernel<<<g_elem64, blk, 0, stream>>>(bufA, sAB2, x_cur,
                                                  N_NODES * DN, DN);
    } else {
      // last layer: x = lrelu(h2) directly (no output BN)
      gemm_fused<2, true, false><<<g_gemm64, blk, 0, stream>>>(
          bufB, nullptr, nullptr, sAB, W2 + (size_t)l * DN * DN,
          b2 + (size_t)l * DN, x_cur, nullptr, N_NODES, DN, DN);
    }
  }

  // ---- classifier head ----
  (void)hipMemsetAsync(pool, 0, (size_t)N_GRAPHS * DN * sizeof(float), stream);
  pool_kernel<<<g_elem64, blk, 0, stream>>>(x_cur, batch, pool, N_NODES * DN);
  hcat_kernel<<<g_elem128, blk, 0, stream>>>(x_cur, pool, batch, bufA,
                                             N_NODES * DM);
  (void)hipMemsetAsync(stats, 0, 2 * DM * sizeof(float), stream);
  gemm_fused<0, false, true><<<g_gemm128, blk, 0, stream>>>(
      bufA, nullptr, nullptr, nullptr, cW1, cb1, bufB, stats, N_NODES, DM, DM);
  bn_finalize<<<1, DM, 0, stream>>>(stats, cbn1g, cbn1b, invn, sAB, DM);
  (void)hipMemsetAsync(stats, 0, 2 * DM * sizeof(float), stream);
  gemm_fused<2, false, true><<<g_gemm128, blk, 0, stream>>>(
      bufB, nullptr, nullptr, sAB, cW2, cb2, bufA, stats, N_NODES, DM, DM);
  bn_finalize<<<1, DM, 0, stream>>>(stats, cbn2g, cbn2b, invn, sAB2, DM);
  final_dot<<<g_dot, blk, 0, stream>>>(bufA, sAB2, cW3, cb3, (float*)d_out,
                                       N_NODES);
}